// MoonLoss_68118181314733
// MI455X (gfx1250) — compile-verified
//
#include <hip/hip_runtime.h>

// MoonLoss on MI455X (gfx1250): pure HBM-bandwidth-bound reduction.
// 335.5 MB of f32 reads -> ~14.4 us floor at 23.3 TB/s. Main kernel is a
// non-temporal b128 streaming pass with deterministic hierarchical reduction;
// the finalize kernel (256 threads) folds the per-block partials with
// vectorized loads and uses V_WMMA_F32_16X16X4_F32 (exact fp32 WMMA) for the
// final 32-lane fold, executed by every wave with EXEC all-1s.

typedef __attribute__((ext_vector_type(4))) float f4;
typedef __attribute__((ext_vector_type(2))) float v2f;
typedef __attribute__((ext_vector_type(8))) float v8f;

#define ML_BLOCKS   4096
#define ML_THREADS  256

// One element: columns 1..4 of F=5 are circular (angles in [0,1)).
__device__ __forceinline__ float ml_sq_err(float o, float t, unsigned int idx) {
    const bool circ = (idx % 5u) != 0u;          // col 0 linear, cols 1..4 circular
    float om = o - floorf(o);                    // jnp.mod(o, 1.0)
    o = circ ? om : o;
    float d = o - t;
    float shift = (fabsf(d) > 0.5f) ? ((t < o) ? 1.0f : -1.0f) : 0.0f;
    t = circ ? (t + shift) : t;
    float e = o - t;
    return e * e;
}

__global__ void moonloss_partial(const float* __restrict__ outs,
                                 const float* __restrict__ tgts,
                                 float* __restrict__ partials,
                                 unsigned int n) {
    const unsigned int n4 = n >> 2;              // float4 count
    const f4* __restrict__ o4 = (const f4*)outs;
    const f4* __restrict__ t4 = (const f4*)tgts;

    float lsum = 0.0f;
    const unsigned int stride = gridDim.x * blockDim.x;
    for (unsigned int i = blockIdx.x * blockDim.x + threadIdx.x; i < n4; i += stride) {
        // Non-temporal: data is streamed exactly once (335 MB > 192 MB L2).
        f4 o = __builtin_nontemporal_load(&o4[i]);
        f4 t = __builtin_nontemporal_load(&t4[i]);
        const unsigned int base = i * 4u;
        lsum += ml_sq_err(o.x, t.x, base + 0u);
        lsum += ml_sq_err(o.y, t.y, base + 1u);
        lsum += ml_sq_err(o.z, t.z, base + 2u);
        lsum += ml_sq_err(o.w, t.w, base + 3u);
    }
    // Tail (n % 4) handled by one thread, deterministically.
    if (blockIdx.x == 0 && threadIdx.x == 0) {
        for (unsigned int i = n4 * 4u; i < n; ++i)
            lsum += ml_sq_err(outs[i], tgts[i], i);
    }

    // wave32 reduction
    #pragma unroll
    for (int off = 16; off > 0; off >>= 1)
        lsum += __shfl_down(lsum, off, 32);

    __shared__ float wsum[ML_THREADS / 32];
    const int lane = threadIdx.x & 31;
    const int wid  = threadIdx.x >> 5;
    if (lane == 0) wsum[wid] = lsum;
    __syncthreads();

    if (wid == 0) {
        float v = (lane < (ML_THREADS / 32)) ? wsum[lane] : 0.0f;
        #pragma unroll
        for (int off = 4; off > 0; off >>= 1)
            v += __shfl_down(v, off, 32);
        if (lane == 0) partials[blockIdx.x] = v;   // plain store: deterministic
    }
}

// 256 threads. Parallel vectorized fold of the per-block partials, then one
// WMMA fold of the 32 lane values, executed by EVERY wave (EXEC all-1s, no
// divergence around the WMMA), with only thread 0 storing the result.
//   A layout (16x4 f32): lane m (0..15) VGPR0 -> A[m][0]; lane m+16 VGPR0 -> A[m][2]
//   With a = {s_lane, 0} and B = all-ones, D[m][*] = s_m + s_{m+16}.
//   Lanes 0..15 hold M=v in VGPR v; lanes 16..31 hold M=v+8.
__global__ void moonloss_final(const float* __restrict__ partials,
                               float* __restrict__ out,
                               unsigned int nparts,
                               float inv_n) {
    const unsigned int tid = threadIdx.x;
    const unsigned int p4n = nparts >> 2;
    const f4* __restrict__ p4 = (const f4*)partials;

    float s = 0.0f;
    for (unsigned int i = tid; i < p4n; i += ML_THREADS) {
        f4 v = p4[i];                     // independent b128 loads, pipelined
        s += (v.x + v.y) + (v.z + v.w);
    }
    if (tid == 0) {                       // tail if nparts % 4 != 0
        for (unsigned int i = p4n * 4u; i < nparts; ++i)
            s += partials[i];
    }

    // wave32 reduction
    #pragma unroll
    for (int off = 16; off > 0; off >>= 1)
        s += __shfl_down(s, off, 32);

    __shared__ float wsum[ML_THREADS / 32];
    const int lane = tid & 31;
    const int wid  = tid >> 5;
    if (lane == 0) wsum[wid] = s;
    __syncthreads();

    // All waves fold the 8 wave-sums redundantly: keeps EXEC all-1s for WMMA.
    float v = (lane < (ML_THREADS / 32)) ? wsum[lane] : 0.0f;

    v2f a; a[0] = v;    a[1] = 0.0f;
    v2f b; b[0] = 1.0f; b[1] = 1.0f;      // all-ones 4x16 B
    v8f c = {0.f, 0.f, 0.f, 0.f, 0.f, 0.f, 0.f, 0.f};
    v8f d = __builtin_amdgcn_wmma_f32_16x16x4_f32(
        /*neg_a=*/false, a, /*neg_b=*/false, b,
        /*c_mod=*/(short)0, c, /*reuse_a=*/false, /*reuse_b=*/false);

    float t = (((d[0] + d[1]) + (d[2] + d[3])) + ((d[4] + d[5]) + (d[6] + d[7])));
    // lane 0 holds sum over M=0..7; lane 16 holds sum over M=8..15.
    float hi = __shfl(t, 16, 32);
    if (tid == 0) out[0] = (t + hi) * inv_n;
}

extern "C" void kernel_launch(void* const* d_in, const int* in_sizes, int n_in,
                              void* d_out, int out_size, void* d_ws, size_t ws_size,
                              hipStream_t stream) {
    const float* outs = (const float*)d_in[0];
    const float* tgts = (const float*)d_in[1];
    float* out = (float*)d_out;
    const unsigned int n = (unsigned int)in_sizes[0];   // B*F elements

    int blocks = ML_BLOCKS;
    if (ws_size < (size_t)blocks * sizeof(float))
        blocks = (int)(ws_size / sizeof(float));
    if (blocks < 1) blocks = 1;

    float* partials = (float*)d_ws;
    const float inv_n = (float)(1.0 / (double)n);

    moonloss_partial<<<blocks, ML_THREADS, 0, stream>>>(outs, tgts, partials, n);
    moonloss_final<<<1, ML_THREADS, 0, stream>>>(partials, out, (unsigned int)blocks, inv_n);
}